// YoloLoss_4784593568428
// MI455X (gfx1250) — compile-verified
//
#include <hip/hip_runtime.h>
#include <hip/hip_bf16.h>
#include <math.h>

#define EPS 1e-7f
#define K4PI2 0.405284734569351f /* 4/pi^2 */

typedef float v2f __attribute__((ext_vector_type(2)));
typedef float v8f __attribute__((ext_vector_type(8)));

// ---------------------------------------------------------------------------
// Per-scale kernel: one thread per prediction cell (b,h,w,anchor).
// grid = (chunks_per_batch, B), block = 256.
// Ground-truth table for this batch row is staged in LDS once per block.
// ---------------------------------------------------------------------------
__global__ void yolo_scale_kernel(
    const float* __restrict__ feats,     // (B,H,W,3*(5+NC))
    const float* __restrict__ ytb,       // (B,T,4)
    const float* __restrict__ iou_mask,  // (B,H,W,3,T)
    const float* __restrict__ cls_true,  // (B,H,W,3,NC)
    int H, int W, int T, int NC, float inv_img,
    float aw0, float ah0, float aw1, float ah1, float aw2, float ah2,
    float* __restrict__ loc_out, float* __restrict__ conf_out,
    float* __restrict__ cls_out)
{
    constexpr int MAXT = 128;
    __shared__ float s_minx[MAXT], s_miny[MAXT], s_maxx[MAXT], s_maxy[MAXT];
    __shared__ float s_cx[MAXT], s_cy[MAXT], s_area[MAXT], s_at[MAXT];
    __shared__ float rloc[256], rconf[256], rcls[256];

    const int tid = threadIdx.x;
    const int b   = blockIdx.y;

    if (tid < T) {
        const float* p = ytb + ((size_t)b * T + tid) * 4;
        float cx = p[0], cy = p[1], wdt = p[2], hgt = p[3];
        s_cx[tid]   = cx;              s_cy[tid]   = cy;
        s_minx[tid] = cx - 0.5f * wdt; s_maxx[tid] = cx + 0.5f * wdt;
        s_miny[tid] = cy - 0.5f * hgt; s_maxy[tid] = cy + 0.5f * hgt;
        s_area[tid] = wdt * hgt;
        s_at[tid]   = atan2f(wdt, hgt);
    }
    __syncthreads();

    const int cells = H * W * 3;
    const int cell  = blockIdx.x * blockDim.x + tid;
    const int FPA   = 5 + NC;          // features per anchor

    float loc = 0.f, conf = 0.f, clsl = 0.f;

    if (cell < cells) {
        int a = cell % 3;
        int w = (cell / 3) % W;
        int h = cell / (3 * W);
        float aw = (a == 0) ? aw0 : ((a == 1) ? aw1 : aw2);
        float ah = (a == 0) ? ah0 : ((a == 1) ? ah1 : ah2);

        const float* f = feats + (size_t)((b * H + h) * W + w) * (3 * FPA)
                               + (size_t)a * FPA;
        float tx = f[0], ty = f[1], tw = f[2], th = f[3], tobj = f[4];

        float px = (1.f / (1.f + expf(-tx)) + (float)w) / (float)W;
        float py = (1.f / (1.f + expf(-ty)) + (float)h) / (float)H;
        float pw = expf(tw) * aw * inv_img;
        float ph = expf(th) * ah * inv_img;

        float pminx = px - 0.5f * pw, pmaxx = px + 0.5f * pw;
        float pminy = py - 0.5f * ph, pmaxy = py + 0.5f * ph;
        float parea = pw * ph;
        float pat   = atan2f(pw, ph);

        const float* mrow = iou_mask +
            (size_t)((((b * H) + h) * W + w) * 3 + a) * T;

        float ciou_sum = 0.f, best = 0.f, obj = 0.f;
        for (int t = 0; t < T; ++t) {
            float iwx = fminf(s_maxx[t], pmaxx) - fmaxf(s_minx[t], pminx);
            float iwy = fminf(s_maxy[t], pmaxy) - fmaxf(s_miny[t], pminy);
            iwx = fmaxf(iwx, 0.f); iwy = fmaxf(iwy, 0.f);
            float inter = iwx * iwy;
            float uni   = s_area[t] + parea - inter;
            float iou   = inter / (uni + EPS);
            float dx = s_cx[t] - px, dy = s_cy[t] - py;
            float center = dx * dx + dy * dy;
            float ewx = fmaxf(fmaxf(s_maxx[t], pmaxx) - fminf(s_minx[t], pminx), 0.f);
            float ewy = fmaxf(fmaxf(s_maxy[t], pmaxy) - fminf(s_miny[t], pminy), 0.f);
            float diag = ewx * ewx + ewy * ewy;
            float diou = iou - center / (diag + EPS);
            float dv = s_at[t] - pat;
            float v  = K4PI2 * dv * dv;
            float alpha = v / (1.f - iou + v + EPS);
            float ciou  = diou - alpha * v;
            float m = mrow[t];
            ciou_sum += (1.f - ciou) * m;
            obj  = fmaxf(obj, m);
            best = fmaxf(best, iou);
        }

        const float* crow = cls_true +
            (size_t)((((b * H) + h) * W + w) * 3 + a) * NC;
        float maxlogit = -INFINITY, csum = 0.f, bsum = 0.f;
        for (int c = 0; c < NC; ++c) {
            float lg = f[5 + c];
            float yt = crow[c];
            maxlogit = fmaxf(maxlogit, lg);
            csum += yt;
            bsum += fmaxf(lg, 0.f) - lg * yt + log1pf(expf(-fabsf(lg)));
        }

        bool  ignore       = best < 0.5f;
        bool  ignore_match = best > 0.5f;
        float maxsig = 1.f / (1.f + expf(-maxlogit));
        bool  cid    = (maxsig < 0.25f) && ignore_match;
        float ignore_mask = (ignore || cid) ? 1.f : 0.f;

        float bce_obj = fmaxf(tobj, 0.f) - tobj * obj + log1pf(expf(-fabsf(tobj)));
        conf = obj * bce_obj + (1.f - obj) * bce_obj * ignore_mask;
        clsl = obj * bsum;
        float cs = csum + ((csum == 0.f) ? 1.f : 0.f);
        loc  = ciou_sum / cs;
    }

    rloc[tid] = loc; rconf[tid] = conf; rcls[tid] = clsl;
    __syncthreads();
    for (int s = 128; s > 0; s >>= 1) {
        if (tid < s) {
            rloc[tid]  += rloc[tid + s];
            rconf[tid] += rconf[tid + s];
            rcls[tid]  += rcls[tid + s];
        }
        __syncthreads();
    }
    if (tid == 0) {
        int idx = b * gridDim.x + blockIdx.x;
        loc_out[idx]  = rloc[0];
        conf_out[idx] = rconf[0];
        cls_out[idx]  = rcls[0];
    }
}

// ---------------------------------------------------------------------------
// WMMA sum: reduce n floats with V_WMMA_F32_16X16X4_F32, A = ones(16x4).
// D[m][n] = sum_k B[k][n] (+C): every D row equals the column sums of B, so
// summing c[0] over the full wave double-counts (rows M=0 and M=8) -> *0.5.
// Must run with EXEC all-ones (full wave, uniform control flow).
// ---------------------------------------------------------------------------
__device__ inline float wmma_sum(const float* __restrict__ p, int n, int lane)
{
    v8f c = {};
    v2f a; a.x = 1.0f; a.y = 1.0f;
    int chunks = (n + 63) >> 6;
    for (int k = 0; k < chunks; ++k) {
        int i0 = k * 64 + lane;
        int i1 = i0 + 32;
        v2f b;
        b.x = (i0 < n) ? p[i0] : 0.0f;
        b.y = (i1 < n) ? p[i1] : 0.0f;
        c = __builtin_amdgcn_wmma_f32_16x16x4_f32(
                /*neg_a=*/false, a, /*neg_b=*/false, b,
                /*c_mod=*/(short)0, c, /*reuse_a=*/false, /*reuse_b=*/false);
    }
    float v = c[0];
    for (int off = 16; off >= 1; off >>= 1)
        v += __shfl_xor(v, off, 32);
    return v * 0.5f;
}

__global__ void yolo_reduce_kernel(const float* __restrict__ ws,
                                   int n0, int n1, float inv_batch,
                                   float* __restrict__ out)
{
    int lane = threadIdx.x;
    const float* loc0  = ws;
    const float* conf0 = ws + n0;
    const float* cls0  = ws + 2 * n0;
    const float* loc1  = ws + 3 * n0;
    const float* conf1 = ws + 3 * n0 + n1;
    const float* cls1  = ws + 3 * n0 + 2 * n1;

    float loc = (wmma_sum(loc0,  n0, lane) + wmma_sum(loc1,  n1, lane)) * inv_batch;
    float cf  = (wmma_sum(conf0, n0, lane) + wmma_sum(conf1, n1, lane)) * inv_batch;
    float cl  = (wmma_sum(cls0,  n0, lane) + wmma_sum(cls1,  n1, lane)) * inv_batch;

    if (lane == 0) {
        out[0] = loc + cf + cl;
        out[1] = loc;
        out[2] = cf;
        out[3] = cl;
    }
}

// ---------------------------------------------------------------------------
static int isqrt_host(int v) {
    int r = 1;
    while ((r + 1) * (r + 1) <= v) ++r;
    return r;
}

extern "C" void kernel_launch(void* const* d_in, const int* in_sizes, int n_in,
                              void* d_out, int out_size, void* d_ws, size_t ws_size,
                              hipStream_t stream)
{
    const float* output0 = (const float*)d_in[0];
    const float* output1 = (const float*)d_in[1];
    const float* ytb     = (const float*)d_in[2];
    const float* mask0   = (const float*)d_in[3];
    const float* mask1   = (const float*)d_in[4];
    const float* class0  = (const float*)d_in[5];
    const float* class1  = (const float*)d_in[6];
    float* out = (float*)d_out;
    float* ws  = (float*)d_ws;

    // Derive shapes:  output0 = B*HW0*3*(5+NC), class0 = B*HW0*3*NC,
    // iou_mask0 = B*HW0*3*T, y_true_box = B*T*4.
    long s0 = in_sizes[0], s5 = in_sizes[5], s3 = in_sizes[3], s2 = in_sizes[2];
    int NC = (int)((5 * s5) / (s0 - s5));          // -> 80
    int T  = (int)((long)NC * s3 / s5);            // -> 100
    int B  = (int)(s2 / (4L * T));                 // -> 64
    int HW0 = (int)(s0 / ((long)B * 3 * (5 + NC)));            // -> 169
    int HW1 = (int)((long)in_sizes[1] / ((long)B * 3 * (5 + NC))); // -> 676
    int H0 = isqrt_host(HW0), W0 = H0;             // 13
    int H1 = isqrt_host(HW1), W1 = H1;             // 26
    float inv_img = 1.0f / (32.0f * (float)H0);    // 1/416 (input_shape from output0)

    int cells0 = H0 * W0 * 3, chunks0 = (cells0 + 255) / 256, n0 = B * chunks0;
    int cells1 = H1 * W1 * 3, chunks1 = (cells1 + 255) / 256, n1 = B * chunks1;

    float* loc0p  = ws;
    float* conf0p = ws + n0;
    float* cls0p  = ws + 2 * n0;
    float* loc1p  = ws + 3 * n0;
    float* conf1p = ws + 3 * n0 + n1;
    float* cls1p  = ws + 3 * n0 + 2 * n1;

    dim3 blk(256);
    // Scale 0 (13x13): ANCHOR_MASK[0] = [3,4,5] -> (81,82),(135,169),(344,319)
    dim3 g0(chunks0, B);
    yolo_scale_kernel<<<g0, blk, 0, stream>>>(
        output0, ytb, mask0, class0, H0, W0, T, NC, inv_img,
        81.f, 82.f, 135.f, 169.f, 344.f, 319.f,
        loc0p, conf0p, cls0p);

    // Scale 1 (26x26): ANCHOR_MASK[1] = [0,1,2] -> (10,14),(23,27),(37,58)
    dim3 g1(chunks1, B);
    yolo_scale_kernel<<<g1, blk, 0, stream>>>(
        output1, ytb, mask1, class1, H1, W1, T, NC, inv_img,
        10.f, 14.f, 23.f, 27.f, 37.f, 58.f,
        loc1p, conf1p, cls1p);

    // Final WMMA-based reduction: one full wave32, uniform control flow.
    yolo_reduce_kernel<<<dim3(1), dim3(32), 0, stream>>>(
        ws, n0, n1, 1.0f / (float)B, out);
}